// StructureAwareAdapter_49563922595873
// MI455X (gfx1250) — compile-verified
//
#include <hip/hip_runtime.h>
#include <hip/hip_bf16.h>

// ---------------- problem constants (match reference) ----------------
#define NN        50000
#define EE        600000
#define LLM_DIM   640
#define HIDDEN    128
#define STRUCT_DIM 16
#define PROJ_OUT  (HIDDEN - STRUCT_DIM)   // 112
#define LN_EPS    1e-5f

typedef __attribute__((ext_vector_type(2))) float v2f;
typedef __attribute__((ext_vector_type(8))) float v8f;
typedef __attribute__((__vector_size__(4 * sizeof(int)))) int v4i_t;

// --------- optional async global->LDS path (guarded; falls back cleanly) ---------
#if __has_builtin(__builtin_amdgcn_global_load_async_to_lds_b128) && \
    __has_builtin(__builtin_amdgcn_s_wait_asynccnt)
#define USE_ASYNC_LDS 1
#else
#define USE_ASYNC_LDS 0
#endif

static __device__ __forceinline__ void tile_copy16(float* lds_dst, const float* gsrc) {
#if USE_ASYNC_LDS
    __builtin_amdgcn_global_load_async_to_lds_b128((v4i_t*)gsrc, (v4i_t*)lds_dst, 0, 0);
#else
    *(float4*)lds_dst = *(const float4*)gsrc;
#endif
}

static __device__ __forceinline__ void tile_copy_wait() {
#if USE_ASYNC_LDS
    __builtin_amdgcn_s_wait_asynccnt(0);
#endif
}

// ---------------- tiny utility kernels ----------------
__global__ void fill_f32(float* __restrict__ p, float v, int n) {
    int i = blockIdx.x * blockDim.x + threadIdx.x;
    if (i < n) p[i] = v;
}

__global__ void deg_accum(const int* __restrict__ dst, float* __restrict__ deg, int e_cnt) {
    int e = blockIdx.x * blockDim.x + threadIdx.x;
    if (e < e_cnt) atomicAdd(&deg[dst[e]], 1.0f);
}

__global__ void rsqrt_k(const float* __restrict__ deg, float* __restrict__ dis, int n) {
    int i = blockIdx.x * blockDim.x + threadIdx.x;
    if (i < n) dis[i] = rsqrtf(deg[i]);
}

// x[:, 112:128] = struct_emb[type_ids]
__global__ void struct_concat(const int* __restrict__ type_ids,
                              const float* __restrict__ emb,
                              float* __restrict__ x) {
    int i = blockIdx.x * blockDim.x + threadIdx.x;   // over N*16
    if (i < NN * STRUCT_DIM) {
        int node = i >> 4;
        int j    = i & 15;
        x[(size_t)node * HIDDEN + PROJ_OUT + j] = emb[type_ids[node] * STRUCT_DIM + j];
    }
}

// ---------------- WMMA f32 GEMM: C[M,NCOLS] = A[M,K] @ B[K,NCOLS] (+bias) ----------------
// Block: 256 threads = 8 wave32. Each wave computes one 16x16 tile.
// Compile-time K / NCOLS / BSTRIDE so all addressing strength-reduces.
// A tile [16 x 64] and B chunk [64 x NCOLS] both staged in LDS.
#define TK 64
#define ASTRIDE 68   // 4*68=272B rows: 16B aligned; banks 4*lrow+ka -> conflict-free

template <int K, int NCOLS, int BSTRIDE>
__global__ void __launch_bounds__(256)
wmma_gemm_f32(const float* __restrict__ A,
              const float* __restrict__ B,
              const float* __restrict__ bias,   // may be nullptr
              float* __restrict__ C) {
    alignas(16) __shared__ float lds_a[16 * ASTRIDE];
    alignas(16) __shared__ float lds_b[TK * BSTRIDE];

    const int tid  = threadIdx.x;
    const int wave = tid >> 5;
    const int lane = tid & 31;
    const int half = lane >> 4;       // 0 -> K pair (0,1); 1 -> K pair (2,3)
    const int lrow = lane & 15;

    const int rowBase = blockIdx.x * 16;
    const int col     = wave * 16 + lrow;
    const bool active = (wave * 16) < NCOLS;        // wave-uniform -> EXEC stays all-ones

    v8f acc = {};

    for (int k0 = 0; k0 < K; k0 += TK) {
        __syncthreads();
        // A tile [16 x TK]: exactly one float4 per thread
        {
            const int r = tid >> 4;          // 0..15
            const int c = (tid & 15) * 4;    // 0..60
            tile_copy16(&lds_a[r * ASTRIDE + c],
                        &A[(size_t)(rowBase + r) * K + (k0 + c)]);
        }
        // B chunk [TK x NCOLS], row-major in LDS with padded stride
        #pragma unroll
        for (int q = tid; q < TK * NCOLS / 4; q += 256) {
            const int k = q / (NCOLS / 4);
            const int c = (q % (NCOLS / 4)) * 4;
            tile_copy16(&lds_b[k * BSTRIDE + c],
                        &B[(size_t)(k0 + k) * NCOLS + c]);
        }
        tile_copy_wait();
        __syncthreads();

        if (active) {
            #pragma unroll
            for (int kk = 0; kk < TK; kk += 4) {
                const int ka = kk + 2 * half;
                v2f a, b;
                a.x = lds_a[lrow * ASTRIDE + ka];
                a.y = lds_a[lrow * ASTRIDE + ka + 1];
                b.x = lds_b[ka * BSTRIDE + col];
                b.y = lds_b[(ka + 1) * BSTRIDE + col];
                acc = __builtin_amdgcn_wmma_f32_16x16x4_f32(
                    /*neg_a=*/false, a, /*neg_b=*/false, b,
                    /*c_mod=*/(short)0, acc, /*reuse_a=*/false, /*reuse_b=*/false);
            }
        }
    }

    if (active) {
        const float bv = bias ? bias[col] : 0.0f;
        #pragma unroll
        for (int j = 0; j < 8; ++j) {
            const int row = rowBase + j + 8 * half;   // VGPR j: M=j (lanes 0-15) / M=j+8 (16-31)
            C[(size_t)row * HIDDEN + col] = acc[j] + bv;
        }
    }
}

// ---------------- edge scatter: agg[dst] += xw[src] * (dis[src]*dis[dst]) ----------------
// one wave per edge; each lane handles 4 consecutive features (float4)
__global__ void edge_scatter(const int* __restrict__ src,
                             const int* __restrict__ dst,
                             const float* __restrict__ dis,
                             const float* __restrict__ xw,
                             float* __restrict__ agg, int e_cnt) {
    const int idx = blockIdx.x * blockDim.x + threadIdx.x;
    const int e   = idx >> 5;
    if (e >= e_cnt) return;
    const int c = (idx & 31) * 4;
    const int s = src[e];
    const int d = dst[e];
    const float norm = dis[s] * dis[d];
    const float4 v = *(const float4*)(xw + (size_t)s * HIDDEN + c);
    float* ap = agg + (size_t)d * HIDDEN + c;
    atomicAdd(ap + 0, v.x * norm);
    atomicAdd(ap + 1, v.y * norm);
    atomicAdd(ap + 2, v.z * norm);
    atomicAdd(ap + 3, v.w * norm);
}

// ---------------- x += relu(agg + xw*deg_inv + b) ----------------
__global__ void combine(float* __restrict__ x,
                        const float* __restrict__ xw,
                        const float* __restrict__ agg,
                        const float* __restrict__ dis,
                        const float* __restrict__ b, int n /* = N*HIDDEN */) {
    int i = blockIdx.x * blockDim.x + threadIdx.x;
    if (i < n) {
        const int node = i >> 7;
        const int h    = i & (HIDDEN - 1);
        const float di = dis[node];
        float v = agg[i] + xw[i] * (di * di) + b[h];
        v = fmaxf(v, 0.0f);
        x[i] += v;
    }
}

// ---------------- LayerNorm over HIDDEN=128, one wave per node ----------------
__global__ void layernorm(const float* __restrict__ x,
                          const float* __restrict__ g,
                          const float* __restrict__ b,
                          float* __restrict__ out) {
    const int node = blockIdx.x * (blockDim.x >> 5) + (threadIdx.x >> 5);
    const int lane = threadIdx.x & 31;
    if (node >= NN) return;
    const int c = lane * 4;
    const float4 v = *(const float4*)(x + (size_t)node * HIDDEN + c);

    float s = v.x + v.y + v.z + v.w;
    #pragma unroll
    for (int o = 16; o > 0; o >>= 1) s += __shfl_xor(s, o, 32);
    const float mean = s * (1.0f / HIDDEN);

    float4 dv = { v.x - mean, v.y - mean, v.z - mean, v.w - mean };
    float vs = dv.x * dv.x + dv.y * dv.y + dv.z * dv.z + dv.w * dv.w;
    #pragma unroll
    for (int o = 16; o > 0; o >>= 1) vs += __shfl_xor(vs, o, 32);
    const float inv = rsqrtf(vs * (1.0f / HIDDEN) + LN_EPS);

    float4 o4;
    o4.x = dv.x * inv * g[c + 0] + b[c + 0];
    o4.y = dv.y * inv * g[c + 1] + b[c + 1];
    o4.z = dv.z * inv * g[c + 2] + b[c + 2];
    o4.w = dv.w * inv * g[c + 3] + b[c + 3];
    *(float4*)(out + (size_t)node * HIDDEN + c) = o4;
}

// ---------------- launch ----------------
extern "C" void kernel_launch(void* const* d_in, const int* in_sizes, int n_in,
                              void* d_out, int out_size, void* d_ws, size_t ws_size,
                              hipStream_t stream) {
    (void)n_in; (void)out_size; (void)ws_size;

    const float* llm      = (const float*)d_in[0];   // [N, 640]
    const int*   type_ids = (const int*)  d_in[1];   // [N]
    const int*   edge     = (const int*)  d_in[2];   // [2, E]
    const float* projW    = (const float*)d_in[3];   // [640, 112]
    const float* projB    = (const float*)d_in[4];   // [112]
    const float* sEmb     = (const float*)d_in[5];   // [5, 16]
    const float* gcnW     = (const float*)d_in[6];   // [2, 128, 128]
    const float* gcnB     = (const float*)d_in[7];   // [2, 128]
    const float* lnG      = (const float*)d_in[8];   // [128]
    const float* lnB      = (const float*)d_in[9];   // [128]
    float*       out      = (float*)d_out;           // [N, 128]

    const int e_cnt = in_sizes[2] / 2;               // 600000

    // workspace layout (floats)
    float* x   = (float*)d_ws;                       // [N,128]
    float* xw  = x   + (size_t)NN * HIDDEN;          // [N,128]
    float* agg = xw  + (size_t)NN * HIDDEN;          // [N,128]
    float* deg = agg + (size_t)NN * HIDDEN;          // [N]
    float* dis = deg + NN;                           // [N]

    const int B256 = 256;

    // degrees with self-loop: deg = 1 + sum over edges into dst
    fill_f32<<<(NN + B256 - 1) / B256, B256, 0, stream>>>(deg, 1.0f, NN);
    deg_accum<<<(e_cnt + B256 - 1) / B256, B256, 0, stream>>>(edge + e_cnt, deg, e_cnt);
    rsqrt_k<<<(NN + B256 - 1) / B256, B256, 0, stream>>>(deg, dis, NN);

    // x[:, :112] = llm @ projW + projB   (WMMA f32), x[:,112:] = struct_emb gather
    // BSTRIDE 120: 2*120 % 64 = 48 -> half-wave bank ranges disjoint; 4*120 % 16 == 0
    wmma_gemm_f32<LLM_DIM, PROJ_OUT, 120><<<NN / 16, B256, 0, stream>>>(llm, projW, projB, x);
    struct_concat<<<(NN * STRUCT_DIM + B256 - 1) / B256, B256, 0, stream>>>(type_ids, sEmb, x);

    // two GCN layers
    for (int l = 0; l < 2; ++l) {
        // BSTRIDE 136: 2*136 % 64 = 16 -> half-wave bank ranges disjoint; 4*136 % 16 == 0
        wmma_gemm_f32<HIDDEN, HIDDEN, 136><<<NN / 16, B256, 0, stream>>>(
            x, gcnW + (size_t)l * HIDDEN * HIDDEN, nullptr, xw);
        (void)hipMemsetAsync(agg, 0, (size_t)NN * HIDDEN * sizeof(float), stream);
        edge_scatter<<<((size_t)e_cnt * 32 + B256 - 1) / B256, B256, 0, stream>>>(
            edge, edge + e_cnt, dis, xw, agg, e_cnt);
        combine<<<(NN * HIDDEN + B256 - 1) / B256, B256, 0, stream>>>(
            x, xw, agg, dis, gcnB + (size_t)l * HIDDEN, NN * HIDDEN);
    }

    // final LayerNorm -> out
    layernorm<<<NN / 8, B256, 0, stream>>>(x, lnG, lnB, out);
}